// TopKRouter_70806830841966
// MI455X (gfx1250) — compile-verified
//
#include <hip/hip_runtime.h>
#include <stdint.h>

typedef __attribute__((ext_vector_type(2))) float v2f;
typedef __attribute__((ext_vector_type(8))) float v8f;

#define T_TOK 8192
#define D_DIM 4096
#define E_EXP 64
#define K_TOP 2
#define CAP   512
#define NSLOT (T_TOK * K_TOP)   // 16384

// ---------------------------------------------------------------------------
// Kernel 1: logits = x @ W^T + b  via V_WMMA_F32_16X16X4_F32
// block = 128 threads (4 waves), each wave owns a 16-row M tile and computes
// all 4 N tiles (64 experts) reusing its A fragment. grid = 8192/64 = 128.
// ---------------------------------------------------------------------------
__global__ __launch_bounds__(128) void router_gemm_kernel(
    const float* __restrict__ x, const float* __restrict__ W,
    const float* __restrict__ bias, float* __restrict__ logits) {
  const int lane = threadIdx.x & 31;
  const int wave = threadIdx.x >> 5;
  const int l15  = lane & 15;        // M (for A) / N (for B,D)
  const int half = lane >> 4;        // K sub-pair selector
  const int m0   = blockIdx.x * 64 + wave * 16;

  // A fragment: lane<16 holds (M=l15, K=k+0,k+1); lane>=16 holds (M=l15, K=k+2,k+3)
  const float* __restrict__ arow  = x + (size_t)(m0 + l15) * D_DIM + half * 2;
  // B fragment (B = W^T, so B[K][N] = W[N][K]): per N-tile nt, expert = nt*16+l15
  const float* __restrict__ brow0 = W + (size_t)(l15 +  0) * D_DIM + half * 2;
  const float* __restrict__ brow1 = W + (size_t)(l15 + 16) * D_DIM + half * 2;
  const float* __restrict__ brow2 = W + (size_t)(l15 + 32) * D_DIM + half * 2;
  const float* __restrict__ brow3 = W + (size_t)(l15 + 48) * D_DIM + half * 2;

  v8f acc0 = {}, acc1 = {}, acc2 = {}, acc3 = {};

  for (int kk = 0; kk < D_DIM; kk += 16) {
    // stream x: prefetch ahead (emits global_prefetch_b8)
    __builtin_prefetch(arow + kk + 1024, 0, 0);
#pragma unroll
    for (int u = 0; u < 16; u += 4) {
      const int k4 = kk + u;
      v2f a  = *(const v2f*)(arow  + k4);
      v2f f0 = *(const v2f*)(brow0 + k4);
      v2f f1 = *(const v2f*)(brow1 + k4);
      v2f f2 = *(const v2f*)(brow2 + k4);
      v2f f3 = *(const v2f*)(brow3 + k4);
      acc0 = __builtin_amdgcn_wmma_f32_16x16x4_f32(false, a, false, f0, (short)0, acc0, false, false);
      acc1 = __builtin_amdgcn_wmma_f32_16x16x4_f32(false, a, false, f1, (short)0, acc1, false, false);
      acc2 = __builtin_amdgcn_wmma_f32_16x16x4_f32(false, a, false, f2, (short)0, acc2, false, false);
      acc3 = __builtin_amdgcn_wmma_f32_16x16x4_f32(false, a, false, f3, (short)0, acc3, false, false);
    }
  }

  // D layout: VGPR v -> M = v + half*8, N = l15 (within each 16-wide N tile)
  const int N = l15;
#pragma unroll
  for (int v = 0; v < 8; ++v) {
    const int M = v + half * 8;
    float* out = logits + (size_t)(m0 + M) * E_EXP + N;
    out[0]  = acc0[v] + bias[N];
    out[16] = acc1[v] + bias[N + 16];
    out[32] = acc2[v] + bias[N + 32];
    out[48] = acc3[v] + bias[N + 48];
  }
}

// ---------------------------------------------------------------------------
// Kernel 2: row softmax over 64 experts (fp32). One wave per row, lane l
// handles columns l and l+32; wave32 shfl_xor reductions.
// ---------------------------------------------------------------------------
__global__ __launch_bounds__(256) void router_softmax_kernel(
    const float* __restrict__ logits, float* __restrict__ probs) {
  const int wave = threadIdx.x >> 5;
  const int lane = threadIdx.x & 31;
  const int row  = blockIdx.x * 8 + wave;
  const float* lr = logits + (size_t)row * E_EXP;
  float v0 = lr[lane];
  float v1 = lr[lane + 32];
  float m = fmaxf(v0, v1);
#pragma unroll
  for (int off = 16; off > 0; off >>= 1) m = fmaxf(m, __shfl_xor(m, off, 32));
  float e0 = expf(v0 - m), e1 = expf(v1 - m);
  float s = e0 + e1;
#pragma unroll
  for (int off = 16; off > 0; off >>= 1) s += __shfl_xor(s, off, 32);
  float inv = 1.0f / s;
  float* pr = probs + (size_t)row * E_EXP;
  pr[lane]      = e0 * inv;
  pr[lane + 32] = e1 * inv;
}

// ---------------------------------------------------------------------------
// Kernel 3: per-token top-2 (ties -> lowest index, matching lax.top_k) and
// normalization. Emits per-slot (expert, unnorm prob, normed prob).
// ---------------------------------------------------------------------------
__global__ __launch_bounds__(256) void router_top2_kernel(
    const float* __restrict__ probs, int* __restrict__ slot_e,
    float* __restrict__ slot_u, float* __restrict__ slot_n) {
  const int t = blockIdx.x * blockDim.x + threadIdx.x;
  if (t >= T_TOK) return;
  const float* p = probs + (size_t)t * E_EXP;
  float p1 = -1.0f; int e1 = 0;
#pragma unroll 8
  for (int e = 0; e < E_EXP; ++e) { float v = p[e]; if (v > p1) { p1 = v; e1 = e; } }
  float p2 = -1.0f; int e2 = 0;
#pragma unroll 8
  for (int e = 0; e < E_EXP; ++e) {
    if (e == e1) continue;
    float v = p[e]; if (v > p2) { p2 = v; e2 = e; }
  }
  const float inv = 1.0f / (p1 + p2);
  slot_e[2 * t]     = e1;  slot_u[2 * t]     = p1;  slot_n[2 * t]     = p1 * inv;
  slot_e[2 * t + 1] = e2;  slot_u[2 * t + 1] = p2;  slot_n[2 * t + 1] = p2 * inv;
}

// ---------------------------------------------------------------------------
// Kernel 4: capacity dispatch. One workgroup per expert: gather matching
// slots into LDS keys (~prob_bits<<32 | flat_idx), bitonic-sort ascending
// (== prob desc, index asc: exact lexsort((-fu, fe)) semantics), keep first
// CAP. Worst case 16384 slots * 8B = 128KB LDS (within CDNA5's 320KB/WGP).
// ---------------------------------------------------------------------------
extern __shared__ unsigned long long lds_keys[];

__global__ __launch_bounds__(1024) void router_dispatch_kernel(
    const int* __restrict__ slot_e, const float* __restrict__ slot_u,
    const float* __restrict__ slot_n, float* __restrict__ expert_probs,
    int* __restrict__ expert_indices) {
  const int e   = blockIdx.x;
  const int tid = threadIdx.x;
  __shared__ int count;
  if (tid == 0) count = 0;
  for (int i = tid; i < NSLOT; i += 1024) lds_keys[i] = ~0ULL;  // sentinel: sorts last
  __syncthreads();

  for (int i = tid; i < NSLOT; i += 1024) {
    if (slot_e[i] == e) {
      unsigned pb = __float_as_uint(slot_u[i]);   // probs > 0 -> bit order == value order
      int pos = atomicAdd(&count, 1);
      lds_keys[pos] = ((unsigned long long)(~pb) << 32) | (unsigned)i;
    }
  }
  __syncthreads();

  // bitonic sort, ascending over full NSLOT
  for (int size = 2; size <= NSLOT; size <<= 1) {
    for (int stride = size >> 1; stride > 0; stride >>= 1) {
      for (int i = tid; i < NSLOT / 2; i += 1024) {
        const int idx     = ((i & ~(stride - 1)) << 1) | (i & (stride - 1));
        const int partner = idx + stride;
        const bool asc = ((idx & size) == 0);
        unsigned long long a = lds_keys[idx];
        unsigned long long b = lds_keys[partner];
        if ((a > b) == asc) { lds_keys[idx] = b; lds_keys[partner] = a; }
      }
      __syncthreads();
    }
  }

  const int cnt = count;
  if (tid < CAP) {
    float pv = 0.0f;
    int   iv = -1;
    if (tid < cnt) {
      unsigned slot = (unsigned)(lds_keys[tid] & 0xFFFFFFFFu);
      pv = slot_n[slot];
      iv = (int)(slot >> 1);  // flat slot -> token id (k == 2)
    }
    expert_probs[(size_t)e * CAP + tid]   = pv;
    expert_indices[(size_t)e * CAP + tid] = iv;
  }
}

// ---------------------------------------------------------------------------
// Host launcher. Inputs: x, padding_mask(unused), W, b, k(=2), cap(=512).
// Output (flat, in return order): logits[T,E] f32 | probs[T,E] f32 |
// expert_probs[E,CAP] f32 | expert_indices[E,CAP] i32 (bit-stored).
// ---------------------------------------------------------------------------
extern "C" void kernel_launch(void* const* d_in, const int* in_sizes, int n_in,
                              void* d_out, int out_size, void* d_ws, size_t ws_size,
                              hipStream_t stream) {
  const float* x = (const float*)d_in[0];
  const float* W = (const float*)d_in[2];
  const float* b = (const float*)d_in[3];

  float* logits = (float*)d_out;
  float* probs  = logits + (size_t)T_TOK * E_EXP;
  float* eprob  = probs  + (size_t)T_TOK * E_EXP;
  int*   eidx   = (int*)(eprob + (size_t)E_EXP * CAP);

  int*   slot_e = (int*)d_ws;
  float* slot_u = (float*)(slot_e + NSLOT);
  float* slot_n = slot_u + NSLOT;

  router_gemm_kernel<<<T_TOK / 64, 128, 0, stream>>>(x, W, b, logits);
  router_softmax_kernel<<<T_TOK / 8, 256, 0, stream>>>(logits, probs);
  router_top2_kernel<<<T_TOK / 256, 256, 0, stream>>>(probs, slot_e, slot_u, slot_n);
  router_dispatch_kernel<<<E_EXP, 1024, NSLOT * sizeof(unsigned long long), stream>>>(
      slot_e, slot_u, slot_n, eprob, eidx);
}